// GSSupervised_49589692399909
// MI455X (gfx1250) — compile-verified
//
#include <hip/hip_runtime.h>
#include <cstdint>
#include <cstddef>

// ---------------------------------------------------------------------------
// GraphSAGE supervised forward for MI455X (gfx1250, wave32).
// Memory-bound pipeline: feats table (102.4 MB) fits in the 192 MB L2, so the
// 282k random row-gathers are L2-resident after first touch. GEMMs (~1.9
// GFLOP) use V_WMMA_F32_16X16X4_F32 to keep exact f32 numerics — matrix
// throughput is nowhere near the bottleneck at this arithmetic intensity.
// Each wave register-blocks a 16x128 C strip (8 accumulators) so the gathered
// A fragment is loaded once per k-step; weights are staged in LDS per block.
// ---------------------------------------------------------------------------

#define N_NODES 200000
#define DIM     128
#define BSEED   1024
#define MAXDEG  128
#define HID     128
#define NCLS    64
#define FAN1    25
#define FAN2    10
#define L1N     (BSEED * FAN1)   /* 25600  */
#define L2N     (L1N * FAN2)     /* 256000 */

typedef __attribute__((ext_vector_type(2))) float v2f;
typedef __attribute__((ext_vector_type(8))) float v8f;

// ----------------------------- Threefry-2x32 -------------------------------
// Matches jax's threefry2x32 rounds. key(42) = {0,42}; fold_in(key,i) =
// threefry(key, {0,i}); random bits for a flat count array are produced as
// threefry(key, x0=i, x1=i+half) with out0 for the low half, out1 for the
// high half. Span 128 is a power of two, so col = bits & 127.
#define TFR(x0, x1, r)                                   \
  {                                                      \
    (x0) += (x1);                                        \
    (x1) = ((x1) << (r)) | ((x1) >> (32 - (r)));         \
    (x1) ^= (x0);                                        \
  }

__device__ __forceinline__ void tf2x32(uint32_t k0, uint32_t k1,
                                       uint32_t x0, uint32_t x1,
                                       uint32_t& o0, uint32_t& o1) {
  const uint32_t ks0 = k0, ks1 = k1, ks2 = k0 ^ k1 ^ 0x1BD11BDAu;
  x0 += ks0; x1 += ks1;
  TFR(x0, x1, 13) TFR(x0, x1, 15) TFR(x0, x1, 26) TFR(x0, x1, 6)
  x0 += ks1; x1 += ks2 + 1u;
  TFR(x0, x1, 17) TFR(x0, x1, 29) TFR(x0, x1, 16) TFR(x0, x1, 24)
  x0 += ks2; x1 += ks0 + 2u;
  TFR(x0, x1, 13) TFR(x0, x1, 15) TFR(x0, x1, 26) TFR(x0, x1, 6)
  x0 += ks0; x1 += ks1 + 3u;
  TFR(x0, x1, 17) TFR(x0, x1, 29) TFR(x0, x1, 16) TFR(x0, x1, 24)
  x0 += ks1; x1 += ks2 + 4u;
  TFR(x0, x1, 13) TFR(x0, x1, 15) TFR(x0, x1, 26) TFR(x0, x1, 6)
  x0 += ks2; x1 += ks0 + 5u;
  o0 = x0; o1 = x1;
}

// One sampled neighbor id per thread.
__global__ __launch_bounds__(256) void sample_kernel(
    const int* __restrict__ src_ids, int n_src, int fan,
    const int* __restrict__ adj, int fold_i, int* __restrict__ out_ids) {
  const int tid   = blockIdx.x * blockDim.x + threadIdx.x;
  const int total = n_src * fan;
  if (tid >= total) return;
  uint32_t fk0, fk1;
  tf2x32(0u, 42u, 0u, (uint32_t)fold_i, fk0, fk1);   // fold_in(key(42), i)
  const int half = (total + 1) >> 1;
  const uint32_t idx = (uint32_t)(tid < half ? tid : tid - half);
  uint32_t o0, o1;
  tf2x32(fk0, fk1, idx, idx + (uint32_t)half, o0, o1);
  const uint32_t bits = (tid < half) ? o0 : o1;
  const int col = (int)(bits & (uint32_t)(MAXDEG - 1));
  const int src = src_ids[tid / fan];
  out_ids[tid] = adj[(size_t)src * MAXDEG + col];
}

// Mean of `fan` gathered 128-wide feature rows; one wave per output row.
// Prefetch the next gathered row (global_prefetch_b8) to hide gather latency.
__global__ __launch_bounds__(256) void gather_mean128(
    const float* __restrict__ feats, const int* __restrict__ ids,
    int n_out, int fan, float* __restrict__ out) {
  const int wave = (int)((blockIdx.x * blockDim.x + threadIdx.x) >> 5);
  const int lane = threadIdx.x & 31;
  if (wave >= n_out) return;
  const int* g = ids + (size_t)wave * fan;
  int id_cur = g[0];
  float4 acc = make_float4(0.f, 0.f, 0.f, 0.f);
  for (int j = 0; j < fan; ++j) {
    const int id_next = (j + 1 < fan) ? g[j + 1] : id_cur;
    __builtin_prefetch(feats + (size_t)id_next * DIM + lane * 4, 0, 3);
    const float4 v = ((const float4*)(feats + (size_t)id_cur * DIM))[lane];
    acc.x += v.x; acc.y += v.y; acc.z += v.z; acc.w += v.w;
    id_cur = id_next;
  }
  const float inv = 1.0f / (float)fan;
  acc.x *= inv; acc.y *= inv; acc.z *= inv; acc.w *= inv;
  ((float4*)(out + (size_t)wave * DIM))[lane] = acc;
}

// Mean of `fan` consecutive 256-wide rows; one wave per output row.
__global__ __launch_bounds__(256) void mean_rows256(
    const float* __restrict__ src, int n_out, int fan,
    float* __restrict__ out) {
  const int wave = (int)((blockIdx.x * blockDim.x + threadIdx.x) >> 5);
  const int lane = threadIdx.x & 31;
  if (wave >= n_out) return;
  const float* base = src + (size_t)wave * fan * 256;
  float4 a0 = make_float4(0.f, 0.f, 0.f, 0.f);
  float4 a1 = make_float4(0.f, 0.f, 0.f, 0.f);
  for (int j = 0; j < fan; ++j) {
    const float4* p = (const float4*)(base + (size_t)j * 256);
    const float4 u = p[lane], w = p[lane + 32];
    a0.x += u.x; a0.y += u.y; a0.z += u.z; a0.w += u.w;
    a1.x += w.x; a1.y += w.y; a1.z += w.z; a1.w += w.w;
  }
  const float inv = 1.0f / (float)fan;
  a0.x *= inv; a0.y *= inv; a0.z *= inv; a0.w *= inv;
  a1.x *= inv; a1.y *= inv; a1.z *= inv; a1.w *= inv;
  float4* o = (float4*)(out + (size_t)wave * 256);
  o[lane] = a0;
  o[lane + 32] = a1;
}

// ---------------------------------------------------------------------------
// Dual GEMM with concat: C[:, 0:128]   = act(A_self @ W_self)  (A rows may be
//                                         gathered via rows_self)
//                        C[:, 128:256] = act(A_agg  @ W_agg)
// blockIdx.y selects the concat half. Each wave owns a 16x128 C strip: the A
// fragment (ISA 32-bit 16x4 layout: lanes 0-15 hold K{0,1}, lanes 16-31 hold
// K{2,3}) is loaded once per k-step and reused across 8 WMMAs into 8 v8f
// accumulators. Weights are staged into LDS in 64-row k-chunks (32 KB),
// shared by the 4 waves of the block; ds reads are bank-conflict-free.
// All selection branches are block-uniform -> EXEC stays all-ones for WMMA.
// ---------------------------------------------------------------------------
#define KCHUNK 64

__global__ __launch_bounds__(128) void wmma_dualgemm(
    const float* __restrict__ A_self, const int* __restrict__ rows_self,
    const float* __restrict__ A_agg,
    const float* __restrict__ W_self, const float* __restrict__ W_agg,
    int K, float* __restrict__ C, int relu) {
  __shared__ float ldsW[KCHUNK * HID];           // 32 KB

  const int wv   = threadIdx.x >> 5;             // 4 waves / block
  const int mt   = blockIdx.x * 4 + wv;          // M tile
  const int half = blockIdx.y;                   // 0 = self, 1 = agg
  const int lane = threadIdx.x & 31;
  const int ml   = lane & 15;
  const bool hi  = lane >= 16;
  const int koff = hi ? 2 : 0;

  const float* A = half ? A_agg : A_self;
  const float* W = half ? W_agg : W_self;

  int row = mt * 16 + ml;
  if (!half && rows_self) row = rows_self[row];
  const float* arow = A + (size_t)row * K;       // lda == K for every operand

  v8f acc[8];
#pragma unroll
  for (int nt = 0; nt < 8; ++nt) acc[nt] = (v8f){};

  for (int kc = 0; kc < K; kc += KCHUNK) {
    __syncthreads();
    // Cooperative stage of W[kc:kc+64, 0:128] -> LDS (2048 float4 / block).
    const float4* wsrc = (const float4*)(W + (size_t)kc * HID);
    for (int t = threadIdx.x; t < (KCHUNK * HID) / 4; t += 128)
      ((float4*)ldsW)[t] = wsrc[t];
    __syncthreads();

    for (int k = 0; k < KCHUNK; k += 4) {
      v2f a;
      a.x = arow[kc + k + koff + 0];
      a.y = arow[kc + k + koff + 1];
#pragma unroll
      for (int nt = 0; nt < 8; ++nt) {
        v2f b;
        b.x = ldsW[(k + koff + 0) * HID + nt * 16 + ml];
        b.y = ldsW[(k + koff + 1) * HID + nt * 16 + ml];
        acc[nt] = __builtin_amdgcn_wmma_f32_16x16x4_f32(
            /*neg_a=*/false, a, /*neg_b=*/false, b,
            /*c_mod=*/(short)0, acc[nt], /*reuse_a=*/false, /*reuse_b=*/false);
      }
    }
  }

  // C/D layout: VGPR r -> M = r (+8 for lanes 16-31), N = lane & 15.
  float* cbase = C + (size_t)(mt * 16 + (hi ? 8 : 0)) * 256 + half * 128 + ml;
#pragma unroll
  for (int nt = 0; nt < 8; ++nt) {
    float* cp = cbase + nt * 16;
#pragma unroll
    for (int r = 0; r < 8; ++r) {
      float v = acc[nt][r];
      if (relu) v = fmaxf(v, 0.0f);
      cp[(size_t)r * 256] = v;
    }
  }
}

// Row L2-normalize (eps 1e-12) then FC: out = (h/||h||) @ fc_w + fc_b.
// One wave per row; 2 output classes per lane; row broadcast via LDS.
__global__ __launch_bounds__(256) void norm_fc(
    const float* __restrict__ h, const float* __restrict__ fcw,
    const float* __restrict__ fcb, int M, float* __restrict__ out) {
  __shared__ float lds[8][256];
  const int wib  = threadIdx.x >> 5;
  const int lane = threadIdx.x & 31;
  const int row  = blockIdx.x * 8 + wib;
  if (row >= M) return;
  const float* x = h + (size_t)row * 256;
  float ss = 0.f;
  for (int j = 0; j < 8; ++j) {
    const float v = x[lane * 8 + j];
    lds[wib][lane * 8 + j] = v;
    ss += v * v;
  }
  for (int off = 16; off > 0; off >>= 1) ss += __shfl_xor(ss, off, 32);
  const float inv = 1.0f / fmaxf(sqrtf(ss), 1e-12f);
  float acc0 = fcb[lane];
  float acc1 = fcb[lane + 32];
  for (int k = 0; k < 256; ++k) {
    const float xv = lds[wib][k] * inv;          // LDS broadcast
    acc0 += xv * fcw[(size_t)k * NCLS + lane];
    acc1 += xv * fcw[(size_t)k * NCLS + lane + 32];
  }
  out[(size_t)row * NCLS + lane]      = acc0;
  out[(size_t)row * NCLS + lane + 32] = acc1;
}

// ---------------------------------------------------------------------------
extern "C" void kernel_launch(void* const* d_in, const int* in_sizes, int n_in,
                              void* d_out, int out_size, void* d_ws,
                              size_t ws_size, hipStream_t stream) {
  (void)in_sizes; (void)n_in; (void)out_size; (void)ws_size;
  const int*   ids   = (const int*)d_in[0];
  const float* feats = (const float*)d_in[1];
  const int*   adj   = (const int*)d_in[2];
  const float* Wx1   = (const float*)d_in[3];
  const float* Wn1   = (const float*)d_in[4];
  const float* Wx2   = (const float*)d_in[5];
  const float* Wn2   = (const float*)d_in[6];
  const float* fcw   = (const float*)d_in[7];
  const float* fcb   = (const float*)d_in[8];
  float* out = (float*)d_out;

  // Workspace layout (~44.2 MB), every region written before it is read.
  char* ws = (char*)d_ws;
  int*   ids1   = (int*)(ws + 0);                       // 25600 ints
  int*   ids2   = (int*)(ws + 102400);                  // 256000 ints
  float* agg_l1 = (float*)(ws + 1126400);               // [25600,128]
  float* agg_l0 = (float*)(ws + 14233600);              // [1024,128]
  float* hs1_l1 = (float*)(ws + 14757888);              // [25600,256]
  float* hs1_l0 = (float*)(ws + 40972288);              // [1024,256]
  float* agg2   = (float*)(ws + 42020864);              // [1024,256]
  float* hs2    = (float*)(ws + 43069440);              // [1024,256]

  // Hop sampling (Threefry).
  sample_kernel<<<(L1N + 255) / 256, 256, 0, stream>>>(ids, BSEED, FAN1, adj, 0, ids1);
  sample_kernel<<<(L2N + 255) / 256, 256, 0, stream>>>(ids1, L1N, FAN2, adj, 1, ids2);

  // Neighbor means (gathers hit L2: feats table < 192 MB L2).
  gather_mean128<<<(L2N / FAN2) / 8, 256, 0, stream>>>(feats, ids2, L1N, FAN2, agg_l1);
  gather_mean128<<<BSEED / 8, 256, 0, stream>>>(feats, ids1, BSEED, FAN1, agg_l0);

  // Layer 1 (relu): level-1 (M=25600) and level-0 (M=1024).
  wmma_dualgemm<<<dim3(L1N / 64, 2), 128, 0, stream>>>(
      feats, ids1, agg_l1, Wx1, Wn1, DIM, hs1_l1, 1);
  wmma_dualgemm<<<dim3(BSEED / 64, 2), 128, 0, stream>>>(
      feats, ids, agg_l0, Wx1, Wn1, DIM, hs1_l0, 1);

  // Mean over 25 level-1 rows -> layer-2 aggregate.
  mean_rows256<<<BSEED / 8, 256, 0, stream>>>(hs1_l1, BSEED, FAN1, agg2);

  // Layer 2 (no act), K=256.
  wmma_dualgemm<<<dim3(BSEED / 64, 2), 128, 0, stream>>>(
      hs1_l0, nullptr, agg2, Wx2, Wn2, 2 * HID, hs2, 0);

  // Normalize + classifier head.
  norm_fc<<<BSEED / 8, 256, 0, stream>>>(hs2, fcw, fcb, BSEED, out);
}